// SpatialLoss_2413771620964
// MI455X (gfx1250) — compile-verified
//
#include <hip/hip_runtime.h>

typedef float v2f __attribute__((ext_vector_type(2)));
typedef float v4f __attribute__((ext_vector_type(4)));
typedef float v8f __attribute__((ext_vector_type(8)));

#define B_   32
#define C_   3
#define H_   512
#define W_   512
#define HP   128
#define WP   128
#define NP   (B_ * HP * WP)   /* 524288 pooled pixels */
#define NBLK (NP / 256)       /* 2048 blocks          */

// ---------------------------------------------------------------------------
// Full-wave (32 lane) f32 sum using V_WMMA_F32_16X16X4_F32.
// A (16x4) holds the lane values: lane L<16 -> A[L,0]=val (A[L,1]=0),
// lane L>=16 -> A[L-16,2]=val (A[L-16,3]=0).  B (4x16) = all ones.
// D[m,n] = sum_k A[m,k] = val[m] + val[m+16].
// Per the 16x16 f32 C/D layout, lane L holds D[m, L%16] for m = 0..7 (+8 for
// the upper half-wave) in d[0..7]; summing d[0..7] gives the half-sum, and one
// shfl_xor(16) completes the 32-lane sum.  Exact f32 RNE — same as a tree add.
// EXEC is all-ones at every call site (full blocks, no divergence here).
// ---------------------------------------------------------------------------
__device__ __forceinline__ float wave_reduce_wmma(float val) {
    v2f a; a[0] = val;  a[1] = 0.0f;
    v2f b; b[0] = 1.0f; b[1] = 1.0f;
    v8f c = {};
    v8f d = __builtin_amdgcn_wmma_f32_16x16x4_f32(
        /*neg_a=*/false, a, /*neg_b=*/false, b,
        /*c_mod=*/(short)0, c, /*reuse_a=*/false, /*reuse_b=*/false);
    float s = ((d[0] + d[1]) + (d[2] + d[3])) + ((d[4] + d[5]) + (d[6] + d[7]));
    s += __shfl_xor(s, 16, 32);
    return s;
}

// ---------------------------------------------------------------------------
// Kernel 1: p[b,i,j] = (sum_{c,4x4 window} x - same for pred) / 48
// One thread per pooled pixel; 24 coalesced 16B non-temporal loads per thread
// (wave issues 512B contiguous per load instruction). 201 MB streamed once.
// ---------------------------------------------------------------------------
__global__ __launch_bounds__(256) void pool_diff_kernel(
    const float* __restrict__ x, const float* __restrict__ pred,
    float* __restrict__ p) {
    const int idx = blockIdx.x * 256 + threadIdx.x;     // < NP
    const int b   = idx >> 14;                          // / (128*128)
    const int rem = idx & 16383;
    const int i   = rem >> 7;
    const int j   = rem & 127;
    const int col  = j << 2;                            // 4j (16B aligned)
    const int row0 = i << 2;                            // 4i

    v4f ax = {0.f, 0.f, 0.f, 0.f};
    v4f ap = {0.f, 0.f, 0.f, 0.f};
#pragma unroll
    for (int c = 0; c < C_; ++c) {
        const int chanBase = ((b * C_ + c) * H_ + row0) * W_ + col;
#pragma unroll
        for (int r = 0; r < 4; ++r) {
            const int off = chanBase + r * W_;
            ax += __builtin_nontemporal_load((const v4f*)(x    + off));
            ap += __builtin_nontemporal_load((const v4f*)(pred + off));
        }
    }
    const v4f dv = ax - ap;
    p[idx] = ((dv[0] + dv[1]) + (dv[2] + dv[3])) * (1.0f / 48.0f);
}

// ---------------------------------------------------------------------------
// Kernel 2: per-pixel gradient-difference energy, block-sum via WMMA wave
// reduction + LDS.  Deterministic per-block partials (no float atomics).
// Zero padding => boundary neighbor contributes 0 (diff = center).
// ---------------------------------------------------------------------------
__global__ __launch_bounds__(256) void grad_loss_kernel(
    const float* __restrict__ p, float* __restrict__ blockSums) {
    const int idx = blockIdx.x * 256 + threadIdx.x;
    const int rem = idx & 16383;
    const int i   = rem >> 7;
    const int j   = rem & 127;

    const float cc = p[idx];
    const float lf = (j > 0)      ? p[idx - 1]  : 0.0f;
    const float rt = (j < WP - 1) ? p[idx + 1]  : 0.0f;
    const float up = (i > 0)      ? p[idx - WP] : 0.0f;
    const float dn = (i < HP - 1) ? p[idx + WP] : 0.0f;

    const float dl = cc - lf, dr = cc - rt, du = cc - up, dd = cc - dn;
    const float v  = dl * dl + dr * dr + du * du + dd * dd;

    const float wsum = wave_reduce_wmma(v);

    __shared__ float lds[8];
    const int lane = threadIdx.x & 31;
    const int wid  = threadIdx.x >> 5;
    if (lane == 0) lds[wid] = wsum;
    __syncthreads();
    if (threadIdx.x == 0) {
        float t = 0.0f;
#pragma unroll
        for (int k = 0; k < 8; ++k) t += lds[k];
        blockSums[blockIdx.x] = t;
    }
}

// ---------------------------------------------------------------------------
// Kernel 3: deterministic final reduction of NBLK partials -> scalar loss.
// ---------------------------------------------------------------------------
__global__ __launch_bounds__(256) void final_reduce_kernel(
    const float* __restrict__ blockSums, float* __restrict__ out) {
    float s = 0.0f;
#pragma unroll
    for (int k = 0; k < NBLK / 256; ++k)
        s += blockSums[k * 256 + threadIdx.x];

    const float wsum = wave_reduce_wmma(s);

    __shared__ float lds[8];
    const int lane = threadIdx.x & 31;
    const int wid  = threadIdx.x >> 5;
    if (lane == 0) lds[wid] = wsum;
    __syncthreads();
    if (threadIdx.x == 0) {
        float t = 0.0f;
#pragma unroll
        for (int k = 0; k < 8; ++k) t += lds[k];
        out[0] = t * (1.0f / (float)NP);   // mean over B*Hp*Wp
    }
}

// ---------------------------------------------------------------------------
// Workspace layout: [0, NP) floats = pooled diff p; [NP, NP+NBLK) = block sums.
// Total ~2.01 MB.
// ---------------------------------------------------------------------------
extern "C" void kernel_launch(void* const* d_in, const int* in_sizes, int n_in,
                              void* d_out, int out_size, void* d_ws, size_t ws_size,
                              hipStream_t stream) {
    (void)in_sizes; (void)n_in; (void)out_size; (void)ws_size;
    const float* x    = (const float*)d_in[0];
    const float* pred = (const float*)d_in[1];
    float* p          = (float*)d_ws;
    float* blockSums  = (float*)d_ws + NP;

    pool_diff_kernel  <<<NBLK, 256, 0, stream>>>(x, pred, p);
    grad_loss_kernel  <<<NBLK, 256, 0, stream>>>(p, blockSums);
    final_reduce_kernel<<<1,   256, 0, stream>>>(blockSums, (float*)d_out);
}